// CustomGraphSAGE_56057913147791
// MI455X (gfx1250) — compile-verified
//
#include <hip/hip_runtime.h>
#include <hip/hip_bf16.h>

#define DD 128
#define XS 132   // LDS row stride for X tile (floats): 2 rows apart -> disjoint banks
#define WS 136   // LDS row stride for W (floats): rows k,k+1 vs k+2,k+3 -> disjoint banks

typedef __attribute__((ext_vector_type(2))) float v2f;
typedef __attribute__((ext_vector_type(8))) float v8f;

// ---------------------------------------------------------------------------
// Degree / normalization
// ---------------------------------------------------------------------------
__global__ void deg_init_kernel(float* __restrict__ deg, int nN) {
    int i = blockIdx.x * blockDim.x + threadIdx.x;
    if (i < nN) deg[i] = 1.0f;                 // self-loop contributes 1
}

__global__ void deg_edge_kernel(const int* __restrict__ dst, float* __restrict__ deg, int nE) {
    int e = blockIdx.x * blockDim.x + threadIdx.x;
    if (e < nE) atomicAdd(&deg[dst[e]], 1.0f);
}

__global__ void dinv_kernel(float* __restrict__ deg, int nN) {
    int i = blockIdx.x * blockDim.x + threadIdx.x;
    if (i < nN) deg[i] = rsqrtf(deg[i]);       // in place: deg -> dinv
}

// ---------------------------------------------------------------------------
// GEMM: H = (relu?)X @ W  via V_WMMA_F32_16X16X4_F32
// Block = 256 threads (8 waves). Block computes 128 rows x 128 cols.
// Wave w computes rows [rowBase + 16w, +16) over all 128 columns:
// 8 accumulator tiles of 16x16, K swept in 32 steps of 4 from LDS.
// ---------------------------------------------------------------------------
__global__ __launch_bounds__(256)
void gcn_gemm_wmma(const float* __restrict__ X, const float* __restrict__ W,
                   float* __restrict__ H, int nN, int applyRelu) {
    __shared__ __align__(16) float sW[DD * WS];       // 128 x 128, stride 136
    __shared__ __align__(16) float sX[128 * XS];      // 128 rows of X, stride 132

    const int tid  = threadIdx.x;
    const int wave = tid >> 5;
    const int lane = tid & 31;
    const int lr   = lane & 15;        // position within half-wave
    const int hi   = lane >> 4;        // 0: lanes 0-15, 1: lanes 16-31
    const int rowBase = blockIdx.x * 128;

    // Stage W into LDS (128x128 floats), float4 per thread per step
    for (int i = tid; i < DD * 32; i += 256) {
        int r  = i >> 5;
        int c4 = (i & 31) << 2;
        float4 w = *(const float4*)(W + r * DD + c4);
        *(float4*)(sW + r * WS + c4) = w;
    }
    // Stage X tile into LDS, with OOB rows zeroed and optional fused ReLU
    for (int i = tid; i < 128 * 32; i += 256) {
        int r  = i >> 5;
        int c4 = (i & 31) << 2;
        int gr = rowBase + r;
        float4 v;
        if (gr < nN) {
            v = *(const float4*)(X + (size_t)gr * DD + c4);
            if (applyRelu) {
                v.x = fmaxf(v.x, 0.0f); v.y = fmaxf(v.y, 0.0f);
                v.z = fmaxf(v.z, 0.0f); v.w = fmaxf(v.w, 0.0f);
            }
        } else {
            v = make_float4(0.0f, 0.0f, 0.0f, 0.0f);
        }
        *(float4*)(sX + r * XS + c4) = v;
    }
    __syncthreads();

    // A-fragment source row for this lane (ISA 16x4 f32 A layout):
    //   lanes 0-15 : M=lane,    VGPR0/1 = K 0/1 of the 4-block
    //   lanes16-31 : M=lane-16, VGPR0/1 = K 2/3 of the 4-block
    const float* xrow = sX + (wave * 16 + lr) * XS;

    v8f acc[8];
#pragma unroll
    for (int t = 0; t < 8; ++t) acc[t] = (v8f){0.f,0.f,0.f,0.f,0.f,0.f,0.f,0.f};

    for (int kb = 0; kb < 32; ++kb) {
        const int k0 = kb * 4 + hi * 2;
        v2f a = *(const v2f*)(xrow + k0);          // contiguous, 8B aligned
        const float* wp = sW + k0 * WS + lr;
#pragma unroll
        for (int t = 0; t < 8; ++t) {
            // B-fragment (4x16): VGPR0 = W[k0][n], VGPR1 = W[k0+1][n]
            v2f b;
            b.x = wp[t * 16];
            b.y = wp[t * 16 + WS];
            acc[t] = __builtin_amdgcn_wmma_f32_16x16x4_f32(
                false, a, false, b, (short)0, acc[t], false, false);
        }
    }

    // C/D layout: VGPR v <-> M = v + 8*hi, N = lr (within 16x16 tile)
    const int row0 = rowBase + wave * 16 + hi * 8;
#pragma unroll
    for (int t = 0; t < 8; ++t) {
        const int col = t * 16 + lr;
#pragma unroll
        for (int v = 0; v < 8; ++v) {
            int r = row0 + v;
            if (r < nN) H[(size_t)r * DD + col] = acc[t][v];
        }
    }
}

// ---------------------------------------------------------------------------
// out[i] = dinv[i]^2 * h[i] + bias      (self-loop term + bias)
// One float4 per thread; 32 threads (one wave-aligned group) per node.
// ---------------------------------------------------------------------------
__global__ void init_out_kernel(const float* __restrict__ h, const float* __restrict__ dinv,
                                const float* __restrict__ bias, float* __restrict__ out, int nN) {
    int idx = blockIdx.x * blockDim.x + threadIdx.x;
    int i = idx >> 5;
    int g = (idx & 31) << 2;
    if (i >= nN) return;
    float di = dinv[i];
    float w  = di * di;
    float4 v = *(const float4*)(h + (size_t)i * DD + g);
    float4 b = *(const float4*)(bias + g);
    float4 r;
    r.x = v.x * w + b.x;  r.y = v.y * w + b.y;
    r.z = v.z * w + b.z;  r.w = v.w * w + b.w;
    *(float4*)(out + (size_t)i * DD + g) = r;
}

// ---------------------------------------------------------------------------
// Edge scatter: out[dst] += dinv[src]*dinv[dst] * h[src]
// 32 consecutive lanes share one edge -> scalar-broadcast index/dinv loads,
// fully coalesced 512B row gather, 4 global_atomic_add_f32 per lane.
// ---------------------------------------------------------------------------
__global__ void edge_scatter_kernel(const float* __restrict__ h, const float* __restrict__ dinv,
                                    const int* __restrict__ src, const int* __restrict__ dst,
                                    float* __restrict__ out, int nE) {
    unsigned idx = blockIdx.x * blockDim.x + threadIdx.x;
    int e = idx >> 5;
    int g = (idx & 31) << 2;
    if (e >= nE) return;
    int s = src[e];
    int d = dst[e];
    float nrm = dinv[s] * dinv[d];
    float4 v = *(const float4*)(h + (size_t)s * DD + g);
    float* o = out + (size_t)d * DD + g;
    atomicAdd(o + 0, nrm * v.x);
    atomicAdd(o + 1, nrm * v.y);
    atomicAdd(o + 2, nrm * v.z);
    atomicAdd(o + 3, nrm * v.w);
}

// ---------------------------------------------------------------------------
extern "C" void kernel_launch(void* const* d_in, const int* in_sizes, int n_in,
                              void* d_out, int out_size, void* d_ws, size_t ws_size,
                              hipStream_t stream) {
    const float* x   = (const float*)d_in[0];
    const int*   ei  = (const int*)d_in[1];     // [2, E] row-major
    const float* W1  = (const float*)d_in[2];
    const float* b1  = (const float*)d_in[3];
    const float* W2  = (const float*)d_in[4];
    const float* b2  = (const float*)d_in[5];
    const float* W3  = (const float*)d_in[6];
    const float* b3  = (const float*)d_in[7];

    const int nN = in_sizes[0] / DD;
    const int nE = in_sizes[1] / 2;
    const int* src = ei;          // edge_index[0]
    const int* dst = ei + nE;     // edge_index[1]

    // Workspace: dinv | bufH (N*D) | bufX (N*D)
    const size_t npad = (size_t)((nN + 127) / 128) * 128;
    float* dinv = (float*)d_ws;
    float* bufH = dinv + npad;
    float* bufX = bufH + npad * DD;
    float* outF = (float*)d_out;

    const int TB = 256;
    const dim3 blk(TB);
    const int gN   = (nN + TB - 1) / TB;
    const int gE   = (nE + TB - 1) / TB;
    const int gNd  = (int)(((size_t)nN * 32 + TB - 1) / TB);
    const int gEd  = (int)(((size_t)nE * 32 + TB - 1) / TB);
    const int gGemm = (nN + 127) / 128;

    // degrees -> dinv (in place)
    deg_init_kernel<<<gN, blk, 0, stream>>>(dinv, nN);
    deg_edge_kernel<<<gE, blk, 0, stream>>>(dst, dinv, nE);
    dinv_kernel<<<gN, blk, 0, stream>>>(dinv, nN);

    // Layer 1
    gcn_gemm_wmma<<<gGemm, blk, 0, stream>>>(x, W1, bufH, nN, 0);
    init_out_kernel<<<gNd, blk, 0, stream>>>(bufH, dinv, b1, bufX, nN);
    edge_scatter_kernel<<<gEd, blk, 0, stream>>>(bufH, dinv, src, dst, bufX, nE);

    // Layer 2 (ReLU of layer-1 output fused into GEMM load)
    gcn_gemm_wmma<<<gGemm, blk, 0, stream>>>(bufX, W2, bufH, nN, 1);
    init_out_kernel<<<gNd, blk, 0, stream>>>(bufH, dinv, b2, bufX, nN);
    edge_scatter_kernel<<<gEd, blk, 0, stream>>>(bufH, dinv, src, dst, bufX, nE);

    // Layer 3 -> d_out
    gcn_gemm_wmma<<<gGemm, blk, 0, stream>>>(bufX, W3, bufH, nN, 1);
    init_out_kernel<<<gNd, blk, 0, stream>>>(bufH, dinv, b3, outF, nN);
    edge_scatter_kernel<<<gEd, blk, 0, stream>>>(bufH, dinv, src, dst, outF, nE);
}